// DeepMDsimpleForces_52871047414483
// MI455X (gfx1250) — compile-verified
//
#include <hip/hip_runtime.h>
#include <math.h>

#define NCELLS 32
#define NPP    16
#define BB     32
#define NN     512        // atoms per sample
#define NFK    500        // Fourier points
#define FFTC_  4
#define LBOX   10.0f
#define K2PI   0.62831853071795864769f   // 2*pi/L
#define SIGMA_ 0.01f

typedef __attribute__((ext_vector_type(2))) float v2f;
typedef __attribute__((ext_vector_type(8))) float v8f;

struct Params {
  const float* Rin;
  const float* p1W[5]; const float* p1b[5];
  const float* p2W[5]; const float* p2b[5];
  const float* plrW[5]; const float* plrb[5];
  const float* fitW[5]; const float* fitb[5];
  const float* flrW[5]; const float* flrb[5];
  const float* linW; const float* linb;
  const float* linlrW; const float* linlrb;
  const float* fftmult;
  float* out;
  float *ws_m, *ws_fpre, *ws_fpim, *ws_fre, *ws_fim, *ws_lrc, *ws_D, *ws_gd;
};

// ---------------- small-net helpers (tanh pyramid, concat-residual) --------
// LDS layout per net: W0@0(2) b0@2(2) W1@4(8) b1@12(4) W2@16(32) b2@48(8)
//                     W3@56(128) b3@184(16) W4@200(512) b4@712(32)  -> 744
template<int DIN>
__device__ __forceinline__ void clayT(const float* W, const float* b,
                                      const float* xi, float* xo) {
  constexpr int DOUT = 2 * DIN;
  #pragma unroll
  for (int j = 0; j < DOUT; ++j) {
    float s = b[j];
    #pragma unroll
    for (int i = 0; i < DIN; ++i) s += xi[i] * W[i * DOUT + j];
    xo[j] = xi[j & (DIN - 1)] + tanhf(s);
  }
}

template<int DIN>
__device__ __forceinline__ void clayTJ(const float* W, const float* b,
                                       const float* xi, const float* ti,
                                       float* xo, float* to) {
  constexpr int DOUT = 2 * DIN;
  #pragma unroll
  for (int j = 0; j < DOUT; ++j) {
    float s = b[j], st = 0.f;
    #pragma unroll
    for (int i = 0; i < DIN; ++i) { float w = W[i * DOUT + j]; s += xi[i] * w; st += ti[i] * w; }
    float y = tanhf(s);
    xo[j] = xi[j & (DIN - 1)] + y;
    to[j] = ti[j & (DIN - 1)] + (1.f - y * y) * st;
  }
}

template<int DIN>
__device__ __forceinline__ void clayR(const float* W, const float* b,
                                      const float* xi, float* xo) {
  constexpr int DOUT = 2 * DIN;
  #pragma unroll
  for (int j = 0; j < DOUT; ++j) {
    float s = b[j];
    #pragma unroll
    for (int i = 0; i < DIN; ++i) s += xi[i] * W[i * DOUT + j];
    xo[j] = xi[j & (DIN - 1)] + fmaxf(s, 0.f);
  }
}

template<int DIN, int DOUT>
__device__ __forceinline__ void layT(const float* W, const float* b,
                                     const float* x, float* y) {
  #pragma unroll
  for (int j = 0; j < DOUT; ++j) {
    float s = b[j];
    #pragma unroll
    for (int i = 0; i < DIN; ++i) s += x[i] * W[i * DOUT + j];
    y[j] = tanhf(s);
  }
}

template<int DIN, int DOUT>
__device__ __forceinline__ void layR(const float* W, const float* b,
                                     const float* x, float* y) {
  #pragma unroll
  for (int j = 0; j < DOUT; ++j) {
    float s = b[j];
    #pragma unroll
    for (int i = 0; i < DIN; ++i) s += x[i] * W[i * DOUT + j];
    y[j] = fmaxf(s, 0.f);
  }
}

__device__ __forceinline__ void net5(const float* w, float x, float o[32]) {
  float x1[2];
  #pragma unroll
  for (int j = 0; j < 2; ++j) x1[j] = tanhf(x * w[j] + w[2 + j]);
  float x2[4];  clayT<2>(w + 4,   w + 12,  x1, x2);
  float x3[8];  clayT<4>(w + 16,  w + 48,  x2, x3);
  float x4[16]; clayT<8>(w + 56,  w + 184, x3, x4);
  clayT<16>(w + 200, w + 712, x4, o);
}

__device__ __forceinline__ void net5_jvp(const float* w, float x, float o[32], float t[32]) {
  float x1[2], t1[2];
  #pragma unroll
  for (int j = 0; j < 2; ++j) {
    float y = tanhf(x * w[j] + w[2 + j]);
    x1[j] = y; t1[j] = (1.f - y * y) * w[j];
  }
  float x2[4],  t2[4];  clayTJ<2>(w + 4,   w + 12,  x1, t1, x2, t2);
  float x3[8],  t3[8];  clayTJ<4>(w + 16,  w + 48,  x2, t2, x3, t3);
  float x4[16], t4[16]; clayTJ<8>(w + 56,  w + 184, x3, t3, x4, t4);
  clayTJ<16>(w + 200, w + 712, x4, t4, o, t);
}

__device__ __forceinline__ void stage_net(float* dst, const float* const* W,
                                          const float* const* B2, int tid, int nt) {
  const int wo[5]  = {0, 4, 16, 56, 200};
  const int wsz[5] = {2, 8, 32, 128, 512};
  const int bo[5]  = {2, 12, 48, 184, 712};
  const int bsz[5] = {2, 4, 8, 16, 32};
  for (int l = 0; l < 5; ++l) {
    for (int i = tid; i < wsz[l]; i += nt) dst[wo[l] + i] = W[l][i];
    for (int i = tid; i < bsz[l]; i += nt) dst[bo[l] + i] = B2[l][i];
  }
}

// fit-net LDS layout: W0@0(1024) b0@1024(16) W1@1040(128) b1@1168(8)
//  W2@1176(32) b2@1208(4) W3@1212(8) b3@1220(2) W4@1222(2) b4@1224(1) -> 1225
#define FW0 0
#define FB0 1024
#define FW1 1040
#define FB1 1168
#define FW2 1176
#define FB2 1208
#define FW3 1212
#define FB3 1220
#define FW4 1222
#define FB4 1224

__device__ __forceinline__ void stage_fit(float* dst, const float* const* W,
                                          const float* const* B2, int tid, int nt) {
  const int wo[5]  = {FW0, FW1, FW2, FW3, FW4};
  const int wsz[5] = {1024, 128, 32, 8, 2};
  const int bo[5]  = {FB0, FB1, FB2, FB3, FB4};
  const int bsz[5] = {16, 8, 4, 2, 1};
  for (int l = 0; l < 5; ++l) {
    for (int i = tid; i < wsz[l]; i += nt) dst[wo[l] + i] = W[l][i];
    for (int i = tid; i < bsz[l]; i += nt) dst[bo[l] + i] = B2[l][i];
  }
}

// ---------------- K0: smeared Fourier multipliers m[c,k] -------------------
__global__ __launch_bounds__(256) void k_prep(Params P) {
  int i = blockIdx.x * 256 + threadIdx.x;
  if (i < FFTC_ * NFK) {
    int j = i % NFK;
    float kv = K2PI * (float)(j - NFK / 2);
    float e = SIGMA_ * kv;
    P.ws_m[i] = P.fftmult[i] * expf(-0.5f * e * e);
  }
}

// ---------------- K1a: partial fRe/fIm sums over 128-atom chunks -----------
__global__ __launch_bounds__(512) void k_fourier(Params P) {
  __shared__ float Rl[128];
  int b = blockIdx.x, q = blockIdx.y;
  for (int i = threadIdx.x; i < 128; i += 512) Rl[i] = P.Rin[b * NN + q * 128 + i];
  __syncthreads();
  int k = threadIdx.x;
  if (k >= NFK) return;
  float kv = K2PI * (float)(k - NFK / 2);
  float sre = 0.f, sim = 0.f;
  for (int n = 0; n < 128; ++n) {
    float s, c; sincosf(Rl[n] * kv, &s, &c);
    sre += c; sim += s;
  }
  size_t o = (size_t)((b << 2) + q) * NFK + k;
  P.ws_fpre[o] = sre;
  P.ws_fpim[o] = sim;
}

// ---------------- K1b: reduce 4 partials -> fRe, fIm -----------------------
__global__ __launch_bounds__(256) void k_fourier2(Params P) {
  int i = blockIdx.x * 256 + threadIdx.x;
  if (i >= BB * NFK) return;
  int b = i / NFK, k = i % NFK;
  float re = 0.f, im = 0.f;
  #pragma unroll
  for (int q = 0; q < 4; ++q) {
    size_t o = (size_t)((b << 2) + q) * NFK + k;
    re += P.ws_fpre[o];
    im += P.ws_fpim[o];
  }
  P.ws_fre[i] = re;
  P.ws_fim[i] = -im;     // fIm = -sum(sin)
}

// ---------------- K2: long-range einsum via WMMA f32 16x16x4 ---------------
// Per wave: 16-atom tile; C(16x16) += A(16x4) * B(4x16) chained over K=500.
// A layout (32-bit 16x4): lanes 0-15 K{0,1} in v0/v1, lanes 16-31 K{2,3}.
// B mirrors A's K ordering; cols = channel (pad 4->16).
__global__ __launch_bounds__(256) void k_lrfeat(Params P) {
  __shared__ float sm[FFTC_ * NFK];
  __shared__ float sre[NFK], sim[NFK];
  int b = blockIdx.x >> 2;                 // 4 blocks (8 tiles each) per sample
  for (int i = threadIdx.x; i < FFTC_ * NFK; i += 256) sm[i] = P.ws_m[i];
  for (int i = threadIdx.x; i < NFK; i += 256) {
    sre[i] = P.ws_fre[b * NFK + i];
    sim[i] = P.ws_fim[b * NFK + i];
  }
  __syncthreads();

  int lane  = threadIdx.x & 31;
  int wv    = threadIdx.x >> 5;
  int tile0 = blockIdx.x * 128 + wv * 16;  // first row (b*N+n) of this tile
  int col   = lane & 15;
  int khalf = lane >> 4;
  float Rrow = P.Rin[tile0 + col];

  int   cm  = (col < 4) ? col : 0;
  float msk = (col < 4) ? 1.f : 0.f;

  v8f acc = {0.f, 0.f, 0.f, 0.f, 0.f, 0.f, 0.f, 0.f};
  for (int c = 0; c < NFK / 4; ++c) {
    int kidx = 4 * c + 2 * khalf;
    float kv0 = K2PI * (float)(kidx - NFK / 2);
    float kv1 = K2PI * (float)(kidx + 1 - NFK / 2);
    float s0, c0, s1, c1;
    sincosf(Rrow * kv0, &s0, &c0);
    sincosf(Rrow * kv1, &s1, &c1);
    v2f a, bv;
    a.x = c0 * sre[kidx]     - s0 * sim[kidx];
    a.y = c1 * sre[kidx + 1] - s1 * sim[kidx + 1];
    bv.x = sm[cm * NFK + kidx]     * msk;
    bv.y = sm[cm * NFK + kidx + 1] * msk;
    acc = __builtin_amdgcn_wmma_f32_16x16x4_f32(false, a, false, bv,
                                                (short)0, acc, false, false);
  }
  if (col < 4) {
    #pragma unroll
    for (int i = 0; i < 8; ++i) {
      int M = i + 8 * khalf;
      P.ws_lrc[(tile0 + M) * 4 + col] = acc[i] * (1.0f / (float)NFK);
    }
  }
}

// ---------------- K3: long-range MLPs (relu pyramids), writes d_out --------
__global__ __launch_bounds__(256) void k_longmlp(Params P) {
  int row = blockIdx.x * 256 + threadIdx.x;
  float in4[4];
  #pragma unroll
  for (int c = 0; c < 4; ++c) in4[c] = P.ws_lrc[row * 4 + c];
  float x1[2];
  #pragma unroll
  for (int j = 0; j < 2; ++j) {
    float s = P.plrb[0][j];
    #pragma unroll
    for (int i = 0; i < 4; ++i) s += in4[i] * P.plrW[0][i * 2 + j];
    x1[j] = fmaxf(s, 0.f);
  }
  float x2[4];  clayR<2>(P.plrW[1], P.plrb[1], x1, x2);
  float x3[8];  clayR<4>(P.plrW[2], P.plrb[2], x2, x3);
  float x4[16]; clayR<8>(P.plrW[3], P.plrb[3], x3, x4);
  float x5[32]; clayR<16>(P.plrW[4], P.plrb[4], x4, x5);
  float h1[16]; layR<32, 16>(P.flrW[0], P.flrb[0], x5, h1);
  float h2[8];  layR<16, 8>(P.flrW[1], P.flrb[1], h1, h2);
  float h3[4];  layR<8, 4>(P.flrW[2], P.flrb[2], h2, h3);
  float h4[2];  layR<4, 2>(P.flrW[3], P.flrb[3], h3, h4);
  float s5 = P.flrb[4][0] + h4[0] * P.flrW[4][0] + h4[1] * P.flrW[4][1];
  float h5 = fmaxf(s5, 0.f);
  P.out[row] = h5 * P.linlrW[0] + P.linlrb[0];
}

// ---------------- K4a: short-range, one WAVE per atom: descriptor D --------
// Lanes own neighbor pairs; per-lane partial D[64] reduced with shfl_xor
// butterfly (deterministic, no atomics).
__global__ __launch_bounds__(256) void k_shortA_pairs(Params P) {
  __shared__ float lds[744 + 744];
  stage_net(lds,       P.p1W, P.p1b, threadIdx.x, 256);
  stage_net(lds + 744, P.p2W, P.p2b, threadIdx.x, 256);
  __syncthreads();
  const float* s1 = lds;
  const float* s2 = lds + 744;

  int lane = threadIdx.x & 31;
  int wv   = threadIdx.x >> 5;
  int row  = blockIdx.x * 8 + wv;                  // one wave per atom row
  int b = row >> 9, n = row & 511, cc = n >> 4, p = n & 15;
  float Rn = P.Rin[row];

  float Dp[64];
  #pragma unroll
  for (int i = 0; i < 64; ++i) Dp[i] = 0.f;

  for (int s = 0; s < 2; ++s) {
    int j = (s == 0) ? lane : 32 + lane;           // 48 neighbor slots
    bool valid = (s == 0) ? (j != NPP + p) : (lane < 16);
    if (valid) {
      int cell = (cc + (j >> 4) + NCELLS - 1) & (NCELLS - 1);
      int mrow = (b << 9) + (cell << 4) + (j & 15);
      float diff = P.Rin[mrow] - Rn;
      diff -= LBOX * rintf(diff * (1.0f / LBOX));  // minimum image
      float r = fabsf(diff), inv = 1.f / r;
      float o[32];
      net5(s1, r, o);
      #pragma unroll
      for (int i = 0; i < 32; ++i) Dp[i] += o[i] * r;        // L1 = P1(r)*r
      net5(s2, inv, o);
      #pragma unroll
      for (int i = 0; i < 32; ++i) Dp[32 + i] += o[i] * inv; // L2 = P2(c0)*c0
    }
  }

  // butterfly-reduce each of the 64 components across the wave
  float keep0 = 0.f, keep1 = 0.f;
  #pragma unroll
  for (int i = 0; i < 64; ++i) {
    float v = Dp[i];
    v += __shfl_xor(v, 1);
    v += __shfl_xor(v, 2);
    v += __shfl_xor(v, 4);
    v += __shfl_xor(v, 8);
    v += __shfl_xor(v, 16);
    if (i < 32) { if (lane == i)        keep0 = v; }
    else        { if (lane == (i - 32)) keep1 = v; }
  }
  float* Dout = P.ws_D + (size_t)row * 64;
  Dout[lane]      = keep0;
  Dout[lane + 32] = keep1;
}

// ---------------- K4b: fit net fwd+bwd per atom -> g_D ---------------------
__global__ __launch_bounds__(256) void k_shortA_fit(Params P) {
  __shared__ float sf[1225];
  stage_fit(sf, P.fitW, P.fitb, threadIdx.x, 256);
  __syncthreads();

  int row = blockIdx.x * 256 + threadIdx.x;
  float D[64];
  const float* Din = P.ws_D + (size_t)row * 64;
  #pragma unroll
  for (int i = 0; i < 64; ++i) D[i] = Din[i];

  float f1[16], f2[8], f3[4], f4[2];
  layT<64, 16>(sf + FW0, sf + FB0, D,  f1);
  layT<16, 8>(sf + FW1, sf + FB1, f1, f2);
  layT<8, 4>(sf + FW2, sf + FB2, f2, f3);
  layT<4, 2>(sf + FW3, sf + FB3, f3, f4);
  float s5 = sf[FB4] + f4[0] * sf[FW4] + f4[1] * sf[FW4 + 1];
  float f5 = tanhf(s5);

  // backward: dE/dF = 1 per row
  float gp5 = P.linW[0] * (1.f - f5 * f5);
  float gp4[2];
  #pragma unroll
  for (int i = 0; i < 2; ++i) gp4[i] = sf[FW4 + i] * gp5 * (1.f - f4[i] * f4[i]);
  float gp3[4];
  #pragma unroll
  for (int i = 0; i < 4; ++i) {
    float g = 0.f;
    #pragma unroll
    for (int j = 0; j < 2; ++j) g += sf[FW3 + i * 2 + j] * gp4[j];
    gp3[i] = g * (1.f - f3[i] * f3[i]);
  }
  float gp2[8];
  #pragma unroll
  for (int i = 0; i < 8; ++i) {
    float g = 0.f;
    #pragma unroll
    for (int j = 0; j < 4; ++j) g += sf[FW2 + i * 4 + j] * gp3[j];
    gp2[i] = g * (1.f - f2[i] * f2[i]);
  }
  float gp1[16];
  #pragma unroll
  for (int i = 0; i < 16; ++i) {
    float g = 0.f;
    #pragma unroll
    for (int j = 0; j < 8; ++j) g += sf[FW1 + i * 8 + j] * gp2[j];
    gp1[i] = g * (1.f - f1[i] * f1[i]);
  }
  float* gOut = P.ws_gd + (size_t)row * 64;
  #pragma unroll
  for (int i = 0; i < 64; ++i) {
    float g = 0.f;
    #pragma unroll
    for (int j = 0; j < 16; ++j) g += sf[FW0 + i * 16 + j] * gp1[j];
    gOut[i] = g;
  }
}

// ---------------- K5: short-range forces, one WAVE per atom ----------------
// Each lane handles 1-2 pairs: JVP of both small nets, dots with g_D[n] (LDS)
// and g_D[m] (global); scalar force butterfly-reduced, lane 0 accumulates.
__global__ __launch_bounds__(256) void k_shortB(Params P) {
  __shared__ float lds[744 + 744];
  __shared__ float gsh[8 * 64];                    // g_D of this block's 8 rows
  stage_net(lds,       P.p1W, P.p1b, threadIdx.x, 256);
  stage_net(lds + 744, P.p2W, P.p2b, threadIdx.x, 256);

  int lane = threadIdx.x & 31;
  int wv   = threadIdx.x >> 5;
  int row  = blockIdx.x * 8 + wv;
  int b = row >> 9, n = row & 511, cc = n >> 4, p = n & 15;

  const float* gsrc = P.ws_gd + (size_t)row * 64;
  gsh[wv * 64 + lane]      = gsrc[lane];
  gsh[wv * 64 + lane + 32] = gsrc[lane + 32];
  __syncthreads();
  const float* s1 = lds;
  const float* s2 = lds + 744;
  const float* gn = gsh + wv * 64;

  float Rn = P.Rin[row];
  float force = 0.f;

  for (int s = 0; s < 2; ++s) {
    int j = (s == 0) ? lane : 32 + lane;
    bool valid = (s == 0) ? (j != NPP + p) : (lane < 16);
    if (valid) {
      int cell = (cc + (j >> 4) + NCELLS - 1) & (NCELLS - 1);
      int mrow = (b << 9) + (cell << 4) + (j & 15);
      float diff = P.Rin[mrow] - Rn;
      diff -= LBOX * rintf(diff * (1.0f / LBOX));
      float r = fabsf(diff);
      float sg = (diff > 0.f) ? 1.f : -1.f;        // dr/dR_m
      float inv = 1.f / r, inv2 = inv * inv;

      float o1[32], t1[32];
      net5_jvp(s1, r, o1, t1);
      #pragma unroll
      for (int i = 0; i < 32; ++i) o1[i] = t1[i] * r + o1[i];    // d(P1(r)*r)/dr
      float o2[32], t2[32];
      net5_jvp(s2, inv, o2, t2);
      #pragma unroll
      for (int i = 0; i < 32; ++i) o2[i] = t2[i] * inv + o2[i];  // d(P2(c0)*c0)/dc0

      const float* gm = P.ws_gd + (size_t)mrow * 64;
      float a1 = 0.f, b1 = 0.f, a2 = 0.f, b2 = 0.f;
      #pragma unroll
      for (int i = 0; i < 32; ++i) { a1 += gn[i] * o1[i]; b1 += gm[i] * o1[i]; }
      #pragma unroll
      for (int i = 0; i < 32; ++i) { a2 += gn[32 + i] * o2[i]; b2 += gm[32 + i] * o2[i]; }
      float dEdr_n = a1 - inv2 * a2;               // dc0/dr = -1/r^2
      float dEdr_m = b1 - inv2 * b2;
      force += sg * (dEdr_n + dEdr_m);             // -dE/dR_n from both roles
    }
  }

  force += __shfl_xor(force, 1);
  force += __shfl_xor(force, 2);
  force += __shfl_xor(force, 4);
  force += __shfl_xor(force, 8);
  force += __shfl_xor(force, 16);
  if (lane == 0) P.out[row] += force;              // add to long-range forces
}

// ---------------- launch ---------------------------------------------------
extern "C" void kernel_launch(void* const* d_in, const int* in_sizes, int n_in,
                              void* d_out, int out_size, void* d_ws, size_t ws_size,
                              hipStream_t stream) {
  (void)in_sizes; (void)n_in; (void)out_size; (void)ws_size;
  Params P;
  int idx = 0;
  P.Rin = (const float*)d_in[idx++];
  for (int l = 0; l < 5; ++l) P.p1W[l]  = (const float*)d_in[idx++];
  for (int l = 0; l < 5; ++l) P.p1b[l]  = (const float*)d_in[idx++];
  for (int l = 0; l < 5; ++l) P.p2W[l]  = (const float*)d_in[idx++];
  for (int l = 0; l < 5; ++l) P.p2b[l]  = (const float*)d_in[idx++];
  for (int l = 0; l < 5; ++l) P.plrW[l] = (const float*)d_in[idx++];
  for (int l = 0; l < 5; ++l) P.plrb[l] = (const float*)d_in[idx++];
  for (int l = 0; l < 5; ++l) P.fitW[l] = (const float*)d_in[idx++];
  for (int l = 0; l < 5; ++l) P.fitb[l] = (const float*)d_in[idx++];
  for (int l = 0; l < 5; ++l) P.flrW[l] = (const float*)d_in[idx++];
  for (int l = 0; l < 5; ++l) P.flrb[l] = (const float*)d_in[idx++];
  P.linW    = (const float*)d_in[idx++];
  P.linb    = (const float*)d_in[idx++];
  P.linlrW  = (const float*)d_in[idx++];
  P.linlrb  = (const float*)d_in[idx++];
  P.fftmult = (const float*)d_in[idx++];
  P.out = (float*)d_out;

  float* w = (float*)d_ws;
  P.ws_m    = w;                 //   2000 (pad 2048)
  P.ws_fpre = w + 2048;          //  64000
  P.ws_fpim = w + 66560;         //  64000
  P.ws_fre  = w + 131072;        //  16000
  P.ws_fim  = w + 147456;        //  16000
  P.ws_lrc  = w + 163840;        //  65536
  P.ws_D    = w + 229376;        // 1048576
  P.ws_gd   = w + 1277952;       // 1048576  (total ~9.3 MB)

  k_prep        <<<8,    256, 0, stream>>>(P);
  k_fourier     <<<dim3(BB, 4), 512, 0, stream>>>(P);  // partial Fourier sums
  k_fourier2    <<<63,   256, 0, stream>>>(P);         // reduce to fRe/fIm
  k_lrfeat      <<<128,  256, 0, stream>>>(P);         // WMMA f32 einsum
  k_longmlp     <<<64,   256, 0, stream>>>(P);         // writes d_out (long)
  k_shortA_pairs<<<2048, 256, 0, stream>>>(P);         // D (wave per atom)
  k_shortA_fit  <<<64,   256, 0, stream>>>(P);         // g_D
  k_shortB      <<<2048, 256, 0, stream>>>(P);         // d_out += short forces
}